// QuantizedLatentDistribution_13348758356123
// MI455X (gfx1250) — compile-verified
//
#include <hip/hip_runtime.h>
#include <hip/hip_bf16.h>

typedef __attribute__((ext_vector_type(2))) float v2f;
typedef __attribute__((ext_vector_type(8))) float v8f;
typedef __attribute__((ext_vector_type(4))) unsigned int v4u;
typedef __attribute__((ext_vector_type(4))) int v4i;
typedef __attribute__((ext_vector_type(8))) int v8i;

#define Bc 128
#define Ac 64
#define Dd 256
#define Kk 2048
#define Mm (Bc * Ac)              // 8192 rows
#define LDA 258                   // padded LDS row stride (bank-conflict free)

// d_out flat layout (reference tuple order)
#define OFF_Q    0                        // quantized_ste (B,A,D)  = 2097152
#define OFF_P    2097152                  // probs (B,A)            = 8192
#define OFF_S    2105344                  // soft_one_hot (B,A,K)   = 16777216
#define OFF_SCAL 18882560                 // [loss, uselessness, diversity]

// ---------------------------------------------------------------------------
// Kernel 1: total_samples = sum(codebook_weights); probs = 1/K; zero scalars
// ---------------------------------------------------------------------------
__global__ __launch_bounds__(256)
void vq_init_kernel(const float* __restrict__ w, float* __restrict__ probs,
                    float* __restrict__ scalars, float* __restrict__ ws_total) {
  __shared__ float red[8];
  const int tid = threadIdx.x;
  float s = 0.f;
  for (int i = tid; i < Kk; i += 256) s += w[i];
  for (int off = 1; off < 32; off <<= 1) s += __shfl_xor(s, off, 32);
  if ((tid & 31) == 0) red[tid >> 5] = s;
  __syncthreads();
  if (tid == 0) {
    float t = 0.f;
    for (int i = 0; i < 8; ++i) t += red[i];
    *ws_total = t;
    scalars[0] = 0.f; scalars[1] = 0.f; scalars[2] = 0.f;
  }
  const float invk = 1.0f / (float)Kk;
  for (int i = tid; i < Mm; i += 256) probs[i] = invk;
}

// ---------------------------------------------------------------------------
// Kernel 2: row squared norms. id<K -> c_sq from codebook, else x_sq of latent
// ---------------------------------------------------------------------------
__global__ __launch_bounds__(256)
void vq_sq_kernel(const float* __restrict__ latent, const float* __restrict__ codebook,
                  float* __restrict__ csq, float* __restrict__ xsq) {
  const int wave = threadIdx.x >> 5, lane = threadIdx.x & 31;
  const int id = blockIdx.x * 8 + wave;                // 0 .. K + M - 1
  const float* src;
  float* dst;
  if (id < Kk) { src = codebook + (size_t)id * Dd; dst = csq + id; }
  else         { src = latent + (size_t)(id - Kk) * Dd; dst = xsq + (id - Kk); }
  float s = 0.f;
  for (int j = lane; j < Dd; j += 32) { float v = src[j]; s += v * v; }
  for (int off = 1; off < 32; off <<= 1) s += __shfl_xor(s, off, 32);
  if (lane == 0) *dst = s;
}

// ---------------------------------------------------------------------------
// Kernel 3: uselessness. One wave per code k. log(|diff|+1)=0 for used codes.
// ---------------------------------------------------------------------------
__global__ __launch_bounds__(256)
void vq_useless_kernel(const float* __restrict__ latent, const float* __restrict__ codebook,
                       const float* __restrict__ weights, const int* __restrict__ rand_idx,
                       const float* __restrict__ ws_total, float* __restrict__ out_ul) {
  const int wave = threadIdx.x >> 5, lane = threadIdx.x & 31;
  const int k = blockIdx.x * 8 + wave;                 // 0..K-1
  const float thresh = (*ws_total) * (1.0f / (100.0f * (float)Kk));
  if (!(weights[k] < thresh)) return;                  // used code -> contributes 0
  const float* rl = latent + (size_t)rand_idx[k] * Dd;
  const float* cb = codebook + (size_t)k * Dd;
  float s = 0.f;
  for (int j = lane; j < Dd; j += 32)
    s += __logf(fabsf(rl[j] - cb[j]) + 1.0f);
  for (int off = 1; off < 32; off <<= 1) s += __shfl_xor(s, off, 32);
  if (lane == 0) atomicAdd(out_ul, s * (1.0f / (float)Kk));
}

// ---------------------------------------------------------------------------
// Kernel 4: latent diversity without the BxB GEMM:
//   ld = 2*S2/(B*A) - 2*sum_{a,d}(sum_b x)^2 / (B*B*A)
// ---------------------------------------------------------------------------
__global__ __launch_bounds__(256)
void vq_div_kernel(const float* __restrict__ latent, float* __restrict__ out_ld) {
  const int idx = blockIdx.x * 256 + threadIdx.x;      // a*D + d, 0..16383
  const float* p = latent + idx;
  float s = 0.f, s2 = 0.f;
  for (int b = 0; b < Bc; ++b) {
    float v = p[(size_t)b * (Ac * Dd)];
    s += v; s2 += v * v;
  }
  float c = s2 * (2.0f / (float)Mm) - s * s * (2.0f / (float)((size_t)Bc * Bc * Ac));
  for (int off = 1; off < 32; off <<= 1) c += __shfl_xor(c, off, 32);
  if ((threadIdx.x & 31) == 0) atomicAdd(out_ld, c);
}

// ---------------------------------------------------------------------------
// Kernel 5 (main): fused distance GEMM (v_wmma_f32_16x16x4_f32) + softmax(-100d)
// + argmin + codebook gather + quantization loss.
// Block = 256 threads = 8 waves; block owns a 16-row M tile and all K=2048 cols.
// A-tile staged by the Tensor Data Mover (pad feature -> LDS stride 258).
// d-outer GEMM loop: 1 ds_load_b64 feeds 16 independent WMMA chains per step.
// ---------------------------------------------------------------------------
__global__ __launch_bounds__(256)
void vq_main_kernel(const float* __restrict__ latent, const float* __restrict__ codebook,
                    const float* __restrict__ csq, const float* __restrict__ xsq,
                    float* __restrict__ out_q, float* __restrict__ out_soft,
                    float* __restrict__ out_loss) {
  __shared__ float ldsA[16 * LDA];
  __shared__ float sh_xsq[16];
  __shared__ float wave_min[8][16];
  __shared__ int   wave_arg[8][16];
  __shared__ float wave_sum[8][16];
  __shared__ float row_min[16];
  __shared__ int   row_arg[16];
  __shared__ float row_inv[16];

  const int tid  = threadIdx.x;
  const int wave = tid >> 5;
  const int lane = tid & 31;
  const int half = lane >> 4;      // 0: rows 0-7, 1: rows 8-15 (WMMA C layout)
  const int l16  = lane & 15;
  const int m0   = blockIdx.x * 16;

#if __has_builtin(__builtin_amdgcn_tensor_load_to_lds) && \
    __has_builtin(__builtin_amdgcn_s_wait_tensorcnt)
  // ---- TDM: DMA the 16x256 f32 latent tile into LDS with hardware row
  // padding (interval=256 DWORDs, amount=2 DWORDs => LDS stride 258). ----
  if (wave == 0) {
    const unsigned long long ga =
        (unsigned long long)(uintptr_t)(latent + (size_t)m0 * Dd);
    const unsigned lds_off = (unsigned)(uintptr_t)(void*)&ldsA[0];
    v4u g0;
    g0.x = 0x1u;                                        // count=1, user mode
    g0.y = lds_off;                                     // lds_addr
    g0.z = (unsigned)(ga & 0xffffffffu);                // global_addr[31:0]
    g0.w = (unsigned)((ga >> 32) & 0x1ffffffu)          // global_addr[56:32]
           | (2u << 30);                                // type = 2 (image)
    v8i g1;
    g1[0] = (2 << 16)      // data_size = 4 bytes
          | (1 << 20)      // pad_enable
          | (7 << 22)      // pad_interval: 256 DWORDs
          | (1 << 25);     // pad_amount:   2 DWORDs
    g1[1] = (int)(256u << 16);        // tensor_dim0[15:0] = 256 (in [31:16])
    g1[2] = (int)(16u << 16);         // tensor_dim0 hi=0 | tensor_dim1 lo = 16
    g1[3] = (int)(256u << 16);        // tensor_dim1 hi=0 | tile_dim0 = 256
    g1[4] = 16;                       // tile_dim1 = 16, tile_dim2 = 0
    g1[5] = 256;                      // tensor_dim0_stride lo32 = 256
    g1[6] = 0;                        // stride0 hi | stride1 lo
    g1[7] = 0;
    const v4i z4 = {0, 0, 0, 0};      // groups 2/3 unused (2D tensor)
    const v8i z8 = {0, 0, 0, 0, 0, 0, 0, 0};
    __builtin_amdgcn_tensor_load_to_lds(g0, g1, z4, z4, z8, 0);
    __builtin_amdgcn_s_wait_tensorcnt(0);
  }
#else
  // ---- fallback: float4 staging ----
  for (int i = tid; i < 16 * 64; i += 256) {
    const int r = i >> 6, c4 = i & 63;
    const float4 v = ((const float4*)(latent + (size_t)(m0 + r) * Dd))[c4];
    float* dst = &ldsA[r * LDA + c4 * 4];
    dst[0] = v.x; dst[1] = v.y; dst[2] = v.z; dst[3] = v.w;
  }
#endif
  if (tid < 16) sh_xsq[tid] = xsq[m0 + tid];
  __syncthreads();

  // ---- distance GEMM, d-outer: one A fragment feeds 16 WMMA chains ----
  v8f acc[16];
#pragma unroll
  for (int t = 0; t < 16; ++t) acc[t] = (v8f){0.f,0.f,0.f,0.f,0.f,0.f,0.f,0.f};

  const float* aptr = &ldsA[l16 * LDA + 2 * half];
  const float* bptr = codebook + (size_t)(wave * 16 + l16) * Dd + 2 * half;
  __builtin_prefetch(bptr, 0, 1);                      // global_prefetch_b8

  for (int d0 = 0; d0 < Dd; d0 += 4) {
    const v2f a = *(const v2f*)(aptr + d0);            // A 16x4 f32 fragment
#pragma unroll
    for (int t = 0; t < 16; ++t) {
      // tile stride: 8 waves * 16 cols = 128 codebook rows = 131072 B (imm24)
      const v2f b = *(const v2f*)(bptr + (size_t)t * (128 * Dd) + d0);
      acc[t] = __builtin_amdgcn_wmma_f32_16x16x4_f32(
          /*neg_a=*/false, a, /*neg_b=*/false, b,
          /*c_mod=*/(short)0, acc[t], /*reuse_a=*/false, /*reuse_b=*/false);
    }
  }

  // ---- epilogue: distances + per-lane running min/argmin ----
  float rmin[8]; int rarg[8];
#pragma unroll
  for (int v = 0; v < 8; ++v) { rmin[v] = 3.4e38f; rarg[v] = 0x7fffffff; }
#pragma unroll
  for (int t = 0; t < 16; ++t) {
    const int n = (t * 8 + wave) * 16 + l16;
    const float cs = csq[n];
#pragma unroll
    for (int v = 0; v < 8; ++v) {
      const float d = sh_xsq[half * 8 + v] + cs - 2.0f * acc[t][v];
      acc[t][v] = d;
      if (d < rmin[v] || (d == rmin[v] && n < rarg[v])) { rmin[v] = d; rarg[v] = n; }
    }
  }

  // ---- per-row min/argmin: half-wave shuffle reduce, then across 8 waves ----
#pragma unroll
  for (int v = 0; v < 8; ++v) {
    float mv = rmin[v]; int av = rarg[v];
    for (int off = 1; off < 16; off <<= 1) {
      float om = __shfl_xor(mv, off, 32);
      int   oa = __shfl_xor(av, off, 32);
      if (om < mv || (om == mv && oa < av)) { mv = om; av = oa; }
    }
    rmin[v] = mv; rarg[v] = av;
  }
  if (l16 == 0) {
#pragma unroll
    for (int v = 0; v < 8; ++v) {
      wave_min[wave][half * 8 + v] = rmin[v];
      wave_arg[wave][half * 8 + v] = rarg[v];
    }
  }
  __syncthreads();
  if (tid < 16) {
    float mv = wave_min[0][tid]; int av = wave_arg[0][tid];
    for (int w = 1; w < 8; ++w) {
      float om = wave_min[w][tid]; int oa = wave_arg[w][tid];
      if (om < mv || (om == mv && oa < av)) { mv = om; av = oa; }
    }
    row_min[tid] = mv; row_arg[tid] = av;
  }
  __syncthreads();

  // ---- exp(-100*(d - rowmin)) and row sums ----
  float rsum[8];
#pragma unroll
  for (int v = 0; v < 8; ++v) rsum[v] = 0.f;
#pragma unroll
  for (int t = 0; t < 16; ++t) {
#pragma unroll
    for (int v = 0; v < 8; ++v) {
      float e = __expf(-100.0f * (acc[t][v] - row_min[half * 8 + v]));
      acc[t][v] = e;
      rsum[v] += e;
    }
  }
#pragma unroll
  for (int v = 0; v < 8; ++v) {
    float s = rsum[v];
    for (int off = 1; off < 16; off <<= 1) s += __shfl_xor(s, off, 32);
    rsum[v] = s;
  }
  if (l16 == 0) {
#pragma unroll
    for (int v = 0; v < 8; ++v) wave_sum[wave][half * 8 + v] = rsum[v];
  }
  __syncthreads();
  if (tid < 16) {
    float s = 0.f;
    for (int w = 0; w < 8; ++w) s += wave_sum[w][tid];
    row_inv[tid] = 1.0f / s;
  }
  __syncthreads();

  // ---- normalized soft_one_hot: the single 67MB global write ----
#pragma unroll
  for (int t = 0; t < 16; ++t) {
    const int col_base = (t * 8 + wave) * 16;
#pragma unroll
    for (int v = 0; v < 8; ++v) {
      const int row = half * 8 + v;
      out_soft[(size_t)(m0 + row) * Kk + col_base + l16] = acc[t][v] * row_inv[row];
    }
  }

  // ---- quantized gather (== quantized_ste) + quantization loss ----
  float lpart = 0.f;
  {
    const int r = tid >> 4;            // row 0..15
    const int dl = tid & 15;           // 16 threads per row
    const int arg = row_arg[r];
    const float* cb = codebook + (size_t)arg * Dd;
    float* q = out_q + (size_t)(m0 + r) * Dd;
#pragma unroll
    for (int j = 0; j < 16; ++j) {
      const int d = dl + j * 16;
      const float cv = cb[d];
      const float xv = ldsA[r * LDA + d];
      q[d] = cv;
      const float df = xv - cv;
      lpart += df * df;
    }
  }
  for (int off = 1; off < 32; off <<= 1) lpart += __shfl_xor(lpart, off, 32);
  if (lane == 0) atomicAdd(out_loss, lpart * (1.0f / (float)Mm));
}

// ---------------------------------------------------------------------------
extern "C" void kernel_launch(void* const* d_in, const int* in_sizes, int n_in,
                              void* d_out, int out_size, void* d_ws, size_t ws_size,
                              hipStream_t stream) {
  (void)in_sizes; (void)n_in; (void)out_size; (void)ws_size;
  const float* latent   = (const float*)d_in[0];   // (B,A,D)
  const float* codebook = (const float*)d_in[1];   // (K,D)
  const float* weights  = (const float*)d_in[2];   // (K,)
  const int*   rand_idx = (const int*)d_in[3];     // (K,)

  float* out   = (float*)d_out;
  float* q     = out + OFF_Q;
  float* probs = out + OFF_P;
  float* soft  = out + OFF_S;
  float* scal  = out + OFF_SCAL;                   // [loss, uselessness, diversity]

  float* ws  = (float*)d_ws;
  float* csq = ws;              // K floats
  float* xsq = ws + Kk;         // M floats
  float* tot = ws + Kk + Mm;    // 1 float

  vq_init_kernel   <<<1,                  256, 0, stream>>>(weights, probs, scal, tot);
  vq_sq_kernel     <<<(Kk + Mm) / 8,      256, 0, stream>>>(latent, codebook, csq, xsq);
  vq_useless_kernel<<<Kk / 8,             256, 0, stream>>>(latent, codebook, weights,
                                                            rand_idx, tot, scal + 1);
  vq_div_kernel    <<<(Ac * Dd) / 256,    256, 0, stream>>>(latent, scal + 2);
  vq_main_kernel   <<<Mm / 16,            256, 0, stream>>>(latent, codebook, csq, xsq,
                                                            q, soft, scal);
}